// QuantumLayer_90924457656774
// MI455X (gfx1250) — compile-verified
//
#include <hip/hip_runtime.h>

typedef __attribute__((ext_vector_type(2))) float v2f;
typedef __attribute__((ext_vector_type(8))) float v8f;

#define NQ 4

// ---------------------------------------------------------------------------
// Kernel 1: compose the full 16x16 complex circuit unitary W from weights
// (3,4,3), then emit the packed real 32x32 GEMM matrix W' into d_ws:
//   final'[b][:] = S'[b][:] @ W'   with S' = [Re(state) | Im(state)]
//   W'[k][j]      =  Re(W[j][k])   (k,j < 16)
//   W'[k][j+16]   =  Im(W[j][k])
//   W'[k+16][j]   = -Im(W[j][k])
//   W'[k+16][j+16]=  Re(W[j][k])
// ---------------------------------------------------------------------------
__global__ __launch_bounds__(256)
void build_unitary_kernel(const float* __restrict__ w, float* __restrict__ Wp)
{
    __shared__ float Rb[2][16][16];
    __shared__ float Ib[2][16][16];
    const int t = threadIdx.x;   // 0..255
    const int k = t & 15;        // row of W (output basis index)
    const int j = t >> 4;        // col of W (input basis index)

    int cur = 0;
    Rb[0][k][j] = (k == j) ? 1.f : 0.f;
    Ib[0][k][j] = 0.f;
    __syncthreads();

    for (int l = 0; l < 3; ++l) {
        // --- Rot(phi, theta, omega) on each wire, M <- G * M ---
        for (int i = 0; i < NQ; ++i) {
            float phi   = w[(l * NQ + i) * 3 + 0];
            float theta = w[(l * NQ + i) * 3 + 1];
            float omega = w[(l * NQ + i) * 3 + 2];
            float sh, ch; sincosf(theta * 0.5f, &sh, &ch);
            float spo, cpo; sincosf(0.5f * (phi + omega), &spo, &cpo);
            float spm, cpm; sincosf(0.5f * (phi - omega), &spm, &cpm);
            float ure[2][2], uim[2][2];
            ure[0][0] =  cpo * ch;  uim[0][0] = -spo * ch;
            ure[0][1] = -cpm * sh;  uim[0][1] = -spm * sh;
            ure[1][0] =  cpm * sh;  uim[1][0] = -spm * sh;
            ure[1][1] =  cpo * ch;  uim[1][1] =  spo * ch;

            const int mask = 1 << (3 - i);          // wire i -> bit (3-i)
            const int b  = (k & mask) ? 1 : 0;
            const int k0 = k & ~mask, k1 = k | mask;
            float r0 = Rb[cur][k0][j], i0 = Ib[cur][k0][j];
            float r1 = Rb[cur][k1][j], i1 = Ib[cur][k1][j];
            float nr = ure[b][0]*r0 - uim[b][0]*i0 + ure[b][1]*r1 - uim[b][1]*i1;
            float ni = ure[b][0]*i0 + uim[b][0]*r0 + ure[b][1]*i1 + uim[b][1]*r1;
            const int nxt = cur ^ 1;
            Rb[nxt][k][j] = nr; Ib[nxt][k][j] = ni;
            __syncthreads();
            cur = nxt;
        }
        // --- CNOT ring, range r = l % (NQ-1) + 1 ---
        const int r = (l % (NQ - 1)) + 1;
        for (int i = 0; i < NQ; ++i) {
            const int c  = i, tg = (i + r) % NQ;
            const int cmask = 1 << (3 - c), tmask = 1 << (3 - tg);
            const int src = (k & cmask) ? (k ^ tmask) : k;
            float nr = Rb[cur][src][j], ni = Ib[cur][src][j];
            const int nxt = cur ^ 1;
            Rb[nxt][k][j] = nr; Ib[nxt][k][j] = ni;
            __syncthreads();
            cur = nxt;
        }
    }

    const float wre = Rb[cur][k][j];   // W[k][j]: j_out = k, k_in = j
    const float wim = Ib[cur][k][j];
    Wp[ j       * 32 + k     ] =  wre;
    Wp[ j       * 32 + k + 16] =  wim;
    Wp[(j + 16) * 32 + k     ] = -wim;
    Wp[(j + 16) * 32 + k + 16] =  wre;
}

// ---------------------------------------------------------------------------
// Kernel 2: state prep + (B x 32) @ (32 x 32) GEMM + (16x16)@(16x4) Z-GEMM,
// all via V_WMMA_F32_16X16X4_F32. 256 threads (8 waves), 256 samples / block.
// ---------------------------------------------------------------------------
#define SSTR 36   // S' row stride (floats): 16B-aligned rows, conflict-free
#define PSTR 18   // prob-staging row stride (floats): conflict-free

__global__ __launch_bounds__(256)
void qstate_wmma_kernel(const float* __restrict__ x, const float* __restrict__ Wp,
                        float* __restrict__ out, int B)
{
    __shared__ float S[256][SSTR];              // S' rows: [Re(16) | Im(16)]
    __shared__ float P[256][PSTR];              // per-sample probability staging
    __shared__ __align__(16) float O[256][4];   // output staging for coalesced stores

    const int t = threadIdx.x;
    const int base = blockIdx.x * 256;
    const int sample = base + t;

    // ---- Phase 1: S'[sample] = [Re(amp_0..15) | Im(amp_0..15)] into LDS ----
    float4 xv = (sample < B) ? ((const float4*)x)[sample]
                             : make_float4(0.f, 0.f, 0.f, 0.f);
    float cc[4], ss[4];
    sincosf(xv.x * 0.5f, &ss[0], &cc[0]);
    sincosf(xv.y * 0.5f, &ss[1], &cc[1]);
    sincosf(xv.z * 0.5f, &ss[2], &cc[2]);
    sincosf(xv.w * 0.5f, &ss[3], &cc[3]);

    float vals[32];
#pragma unroll
    for (int k = 0; k < 16; ++k) {
        float m = (((k >> 3) & 1) ? ss[0] : cc[0])
                * (((k >> 2) & 1) ? ss[1] : cc[1])
                * (((k >> 1) & 1) ? ss[2] : cc[2])
                * (( k       & 1) ? ss[3] : cc[3]);
        const int pc = __popc(k) & 3;            // amp = m * (-i)^popcount
        vals[k]      = (pc == 0) ? m : ((pc == 2) ? -m : 0.f);   // Re
        vals[16 + k] = (pc == 1) ? -m : ((pc == 3) ? m : 0.f);   // Im
    }
    float4* srow = (float4*)&S[t][0];            // row start is 16B aligned
#pragma unroll
    for (int i = 0; i < 8; ++i)
        srow[i] = make_float4(vals[4*i], vals[4*i+1], vals[4*i+2], vals[4*i+3]);
    __syncthreads();

    const int lane = t & 31;
    const int wav  = t >> 5;
    const int nlo  = lane & 15;
    const int hi2  = (lane >> 4) << 1;           // 0 (lanes 0-15) or 2 (16-31)
    const int hi8  = (lane >> 4) << 3;           // 0 or 8

    // ---- Preload W' B-fragments (4x16 f32: VGPR0=row hi2, VGPR1=row hi2+1) --
    v2f bf[16];
#pragma unroll
    for (int kk = 0; kk < 8; ++kk) {
#pragma unroll
        for (int n = 0; n < 2; ++n) {
            const int row0 = kk * 4 + hi2;
            const int col  = n * 16 + nlo;
            v2f b;
            b.x = Wp[ row0      * 32 + col];
            b.y = Wp[(row0 + 1) * 32 + col];
            bf[kk * 2 + n] = b;
        }
    }

    // ---- Constant Z B-fragments: Z[j][w] = (bit_(3-w)(j)) ? -1 : +1, w<4 ----
    v2f zf[4];
#pragma unroll
    for (int kk = 0; kk < 4; ++kk) {
        const int j0 = kk * 4 + hi2;
        const int j1 = j0 + 1;
        float z0 = 0.f, z1 = 0.f;
        if (nlo < 4) {
            z0 = ((j0 >> (3 - nlo)) & 1) ? -1.f : 1.f;
            z1 = ((j1 >> (3 - nlo)) & 1) ? -1.f : 1.f;
        }
        zf[kk].x = z0; zf[kk].y = z1;
    }

    // ---- Phase 2: each wave runs two 16-sample tiles ----
#pragma unroll
    for (int tloc = 0; tloc < 2; ++tloc) {
        const int tile = wav * 2 + tloc;
        const int arow = tile * 16 + nlo;        // A: lane holds sample row M

        v8f c0 = {};   // cols  0..15 -> Re(final)
        v8f c1 = {};   // cols 16..31 -> Im(final)
#pragma unroll
        for (int kk = 0; kk < 8; ++kk) {
            const int acol = kk * 4 + hi2;       // even -> 8B-aligned ds_load_b64
            v2f a = *(const v2f*)&S[arow][acol];
            c0 = __builtin_amdgcn_wmma_f32_16x16x4_f32(
                     false, a, false, bf[kk * 2 + 0], (short)0, c0, false, false);
            c1 = __builtin_amdgcn_wmma_f32_16x16x4_f32(
                     false, a, false, bf[kk * 2 + 1], (short)0, c1, false, false);
        }

        // ---- probs -> LDS staging (C-layout: VGPR v -> M=v|v+8, N=lane%16) --
#pragma unroll
        for (int v = 0; v < 8; ++v) {
            const float p = c0[v] * c0[v] + c1[v] * c1[v];
            P[tile * 16 + v + hi8][nlo] = p;
        }
        // Wave-private LDS RAW (each wave reads only rows it wrote): a dscnt
        // wait + compiler barrier is sufficient, no workgroup barrier needed.
        __asm__ volatile("s_wait_dscnt 0" ::: "memory");

        // ---- Z-measurement GEMM: D[M][w] = sum_j P[M][j] * Z[j][w] ----
        v8f d = {};
#pragma unroll
        for (int kk = 0; kk < 4; ++kk) {
            const int pcol = kk * 4 + hi2;       // even -> 8B-aligned ds_load_b64
            v2f a = *(const v2f*)&P[tile * 16 + nlo][pcol];
            d = __builtin_amdgcn_wmma_f32_16x16x4_f32(
                    false, a, false, zf[kk], (short)0, d, false, false);
        }

        // ---- Stage out columns 0..3 (lane N=w<4 holds out[M][w]) ----
        if (nlo < 4) {
#pragma unroll
            for (int v = 0; v < 8; ++v)
                O[tile * 16 + v + hi8][nlo] = d[v];
        }
        __asm__ volatile("s_wait_dscnt 0" ::: "memory");

        // ---- One fully-coalesced 256B store per tile: 32 lanes x float2 ----
        {
            const int gs = base + tile * 16 + (lane >> 1);   // sample for this lane
            if (gs < B) {
                float2 o2 = ((const float2*)&O[tile * 16][0])[lane];
                ((float2*)out)[(size_t)(base + tile * 16) * 2 + lane] = o2;
            }
        }
    }
}

// ---------------------------------------------------------------------------
extern "C" void kernel_launch(void* const* d_in, const int* in_sizes, int n_in,
                              void* d_out, int out_size, void* d_ws, size_t ws_size,
                              hipStream_t stream) {
    const float* x = (const float*)d_in[0];   // (B, 4) f32
    const float* w = (const float*)d_in[1];   // (3, 4, 3) f32
    float* out = (float*)d_out;               // (B, 4) f32
    float* Wp  = (float*)d_ws;                // 32*32 f32 packed unitary

    const int B = in_sizes[0] / 4;

    build_unitary_kernel<<<1, 256, 0, stream>>>(w, Wp);
    const int blocks = (B + 255) / 256;
    qstate_wmma_kernel<<<blocks, 256, 0, stream>>>(x, Wp, out, B);
}